// StructureLoss_70523363000961
// MI455X (gfx1250) — compile-verified
//
#include <hip/hip_runtime.h>
#include <hip/hip_bf16.h>
#include <math.h>

// ---------------------------------------------------------------------------
// StructureLoss (3D SSIM-style) fused kernel for gfx1250 (MI455X).
//
// Simplifications (faithful to reference):
//  * yr = one_hot(y)  => yr*yr = yr, so uyy == uy  (4 box statistics, not 5)
//  * data_range = max(yr) = 1 always (labels in [0,4) => one-hot has a 1)
//    => c = (0.03*1)^2 / 2 = 4.5e-4 constant.
//  * xr = sigmoid(x*m): m==1 -> sigmoid(x); m==0 -> 0.5
//
// 7-tap 'valid' box sums along W and H (22 in -> 16 out rows) are cast as
// D = Band(16x22) * In(22xN) and run on V_WMMA_F32_16X16X4_F32 (6 k-chunks).
// The contiguous D axis is summed in registers while streaming global loads.
// Rows 22/23 of the padded band are handled by clamping the LDS row index to
// 21 (A is 0 there), keeping EXEC uniform around the WMMA ops.
// ---------------------------------------------------------------------------

typedef float v2f __attribute__((ext_vector_type(2)));
typedef float v8f __attribute__((ext_vector_type(8)));

#define HDIM   160
#define WDIM   160
#define DDIM   96
#define NCLS   4
#define OH     154   // 160 - 6
#define OD     90    // 96 - 6
#define TILE   16
#define HLEN   22    // TILE + WIN - 1
#define NSTAT  4
#define DT_TILES 6   // ceil(90/16)
#define WT_TILES 10  // ceil(154/16)
#define HT_TILES 10
#define NBLOCKS (DT_TILES * WT_TILES * HT_TILES * 2)  // 1200
#define NVALID  17075520.0f                            // 8*154*154*90

__device__ __forceinline__ int imin(int a, int b) { return a < b ? a : b; }

__device__ __forceinline__ float fast_sigmoid(float v) {
    return __builtin_amdgcn_rcpf(1.0f + __expf(-v));
}

// Banded 16x22 (7-wide band of ones) times In(22 x cols) for one 16-wide
// column group. Each lane handles column `col`; returns the 16x16 f32 D tile
// in the ISA accumulator layout (lanes 0-15: M=r, lanes 16-31: M=8+r).
// Rows >= 22 are clamped to row 21; the matching A entries are 0, so the
// loads are unconditional (no EXEC divergence) and contribute nothing.
__device__ __forceinline__ v8f band_colsum(const float* __restrict__ Sin,
                                           int stride, int col,
                                           const v2f* __restrict__ A,
                                           int half, int li) {
#if __has_builtin(__builtin_amdgcn_wmma_f32_16x16x4_f32)
    v8f d = {};
    #pragma unroll
    for (int k = 0; k < 6; ++k) {
        const int r0 = imin(4 * k + 2 * half, HLEN - 1);
        const int r1 = imin(4 * k + 2 * half + 1, HLEN - 1);
        v2f bv;
        bv.x = Sin[r0 * stride + col];
        bv.y = Sin[r1 * stride + col];
        d = __builtin_amdgcn_wmma_f32_16x16x4_f32(
                /*neg_a=*/false, A[k], /*neg_b=*/false, bv,
                /*c_mod=*/(short)0, d, /*reuse_a=*/false, /*reuse_b=*/false);
    }
    return d;
#else
    v8f d;
    #pragma unroll
    for (int r = 0; r < 8; ++r) {
        const int m = r + 8 * half;  // output row
        float s = 0.0f;
        #pragma unroll
        for (int k = 0; k < 7; ++k) s += Sin[(m + k) * stride + col];
        d[r] = s;
    }
    return d;
#endif
}

__launch_bounds__(256, 1)
__global__ void structure_loss_tile(const float* __restrict__ x,
                                    const int* __restrict__ y,
                                    float* __restrict__ partial) {
    // [h][w][d'][stat]  (h,w are halo coords 0..21, d' output 0..15)
    __shared__ float S1[HLEN * HLEN * TILE * NSTAT];   // 123,904 B
    // [h][w'][d'][stat]
    __shared__ float S2[HLEN * TILE * TILE * NSTAT];   //  90,112 B
    __shared__ float red[8];
    float* const S3 = S1;  // [h'][w'][d'][stat] aliases S1 (S1 dead by then)

    const int tid  = threadIdx.x;
    const int lane = tid & 31;
    const int wave = tid >> 5;
    const int li   = lane & 15;
    const int half = lane >> 4;

    const int dt = blockIdx.x % DT_TILES;
    const int wt = blockIdx.x / DT_TILES;
    const int ht = blockIdx.y;
    const int b  = blockIdx.z;
    const int h0 = ht * TILE, w0 = wt * TILE, d0 = dt * TILE;
    const bool dfast = (d0 + 24 <= DDIM);   // block-uniform: no d-clamp needed

    // Constant band matrix A (band of 7 ones per row), 6 k-chunks of 4.
    // ISA 16x4 f32 A layout: lane L<16 -> M=L, VGPR0=K0,VGPR1=K1;
    //                        lane L>=16 -> M=L-16, VGPR0=K2,VGPR1=K3.
    v2f A[6];
    #pragma unroll
    for (int k = 0; k < 6; ++k) {
        const int c0 = 4 * k + 2 * half;
        const int c1 = c0 + 1;
        A[k].x = (c0 >= li && c0 <= li + 6) ? 1.0f : 0.0f;
        A[k].y = (c1 >= li && c1 <= li + 6) ? 1.0f : 0.0f;
    }

    const float inv343 = 1.0f / 343.0f;
    const float covn   = 49.0f / 48.0f;     // WIN^2 / (WIN^2 - 1)
    const float cstab  = 0.00045f;          // (0.03*1)^2 / 2

    float acc = 0.0f;

    for (int cls = 0; cls < NCLS; ++cls) {
        // ---- Stage 1: stats + sliding 7-sum along contiguous D axis ----
        for (int p = tid; p < HLEN * HLEN; p += 256) {
            const int hh = p / HLEN, ww = p % HLEN;
            const int gh = imin(h0 + hh, HDIM - 1);
            const int gw = imin(w0 + ww, WDIM - 1);
            const float* xrow = x + (((b * NCLS + cls) * HDIM + gh) * WDIM + gw) * DDIM;
            const int*   yrow = y + ((b * HDIM + gh) * WDIM + gw) * DDIM;

            float xv[24];
            unsigned mbits = 0;
            if (dfast) {
                // 16B-aligned vector loads (row base % 384B == 0, d0 % 64B == 0)
                #pragma unroll
                for (int q = 0; q < 6; ++q) {
                    const float4 xf = ((const float4*)(xrow + d0))[q];
                    const int4   yi = ((const int4*)(yrow + d0))[q];
                    xv[4 * q + 0] = xf.x; xv[4 * q + 1] = xf.y;
                    xv[4 * q + 2] = xf.z; xv[4 * q + 3] = xf.w;
                    mbits |= (unsigned)(yi.x == cls) << (4 * q + 0);
                    mbits |= (unsigned)(yi.y == cls) << (4 * q + 1);
                    mbits |= (unsigned)(yi.z == cls) << (4 * q + 2);
                    mbits |= (unsigned)(yi.w == cls) << (4 * q + 3);
                }
            } else {
                #pragma unroll
                for (int k = 0; k < HLEN; ++k) {
                    const int dd = imin(d0 + k, DDIM - 1);
                    xv[k] = xrow[dd];
                    mbits |= (unsigned)(yrow[dd] == cls) << k;
                }
            }

            float xr[HLEN];
            #pragma unroll
            for (int k = 0; k < HLEN; ++k)
                xr[k] = ((mbits >> k) & 1u) ? fast_sigmoid(xv[k]) : 0.5f;

            float s0 = 0, s1 = 0, s2 = 0, s3 = 0;
            #pragma unroll
            for (int k = 0; k < 7; ++k) {
                const float v  = xr[k];
                const float mf = (float)((mbits >> k) & 1u);
                s0 += v; s1 += v * v; s2 += mf; s3 += mf * v;
            }
            float* out = &S1[p * (TILE * NSTAT)];
            out[0] = s0; out[1] = s1; out[2] = s2; out[3] = s3;
            #pragma unroll
            for (int dp = 1; dp < TILE; ++dp) {
                const int ka = dp + 6, kb = dp - 1;
                const float va = xr[ka], vb = xr[kb];
                const float ma = (float)((mbits >> ka) & 1u);
                const float mb = (float)((mbits >> kb) & 1u);
                s0 += va - vb;
                s1 += va * va - vb * vb;
                s2 += ma - mb;
                s3 += ma * va - mb * vb;
                out[dp * 4 + 0] = s0; out[dp * 4 + 1] = s1;
                out[dp * 4 + 2] = s2; out[dp * 4 + 3] = s3;
            }
        }
        __syncthreads();

        // ---- Stage 2: 7-sum along W via WMMA band-matmul ----
        // 88 (h, colgroup) tiles over 8 waves, uniform per wave (EXEC all 1s).
        for (int t = 0; t < 11; ++t) {
            const int idx = wave + t * 8;     // 0..87
            const int h = idx >> 2, g = idx & 3;
            const int col = g * 16 + li;      // column index in (d',stat)=64
            const v8f d = band_colsum(&S1[h * HLEN * 64], 64, col, A, half, li);
            #pragma unroll
            for (int r = 0; r < 8; ++r) {
                const int wp = r + 8 * half;  // output w'
                S2[(h * TILE + wp) * 64 + col] = d[r];
            }
        }
        __syncthreads();

        // ---- Stage 3: 7-sum along H via WMMA band-matmul ----
        // S2 viewed as 22 rows x 1024 cols (col = w'*64 + d'*4 + stat).
        for (int t = 0; t < 8; ++t) {
            const int g = wave * 8 + t;       // 0..63
            const int col = g * 16 + li;      // 0..1023
            const v8f d = band_colsum(S2, 1024, col, A, half, li);
            #pragma unroll
            for (int r = 0; r < 8; ++r) {
                const int hp = r + 8 * half;  // output h'
                S3[hp * 1024 + col] = d[r];
            }
        }
        __syncthreads();

        // ---- Stage 4: SSIM combine + masked accumulate ----
        for (int v = tid; v < TILE * TILE * TILE; v += 256) {
            const int hp = v >> 8;
            const int wp = (v >> 4) & 15;
            const int dp = v & 15;
            const float4 f = *(const float4*)&S3[hp * 1024 + wp * 64 + dp * 4];
            const float ux  = f.x * inv343;
            const float uxx = f.y * inv343;
            const float uy  = f.z * inv343;   // == uyy (binary mask)
            const float uxy = f.w * inv343;
            const float vx  = covn * (uxx - ux * ux);
            const float vy  = covn * (uy - uy * uy);
            const float vxy = covn * (uxy - ux * uy);
            const float sv  = (vxy + cstab) *
                              __builtin_amdgcn_rcpf(vx * vy + cstab);
            const bool ok = (h0 + hp < OH) && (w0 + wp < OH) && (d0 + dp < OD);
            acc += ok ? sv : 0.0f;
        }
        __syncthreads();  // S3 (=S1) must be consumed before next class
    }

    // ---- Deterministic block reduction ----
    #pragma unroll
    for (int off = 16; off > 0; off >>= 1)
        acc += __shfl_down(acc, off, 32);
    if (lane == 0) red[wave] = acc;
    __syncthreads();
    if (tid == 0) {
        float t = 0.0f;
        #pragma unroll
        for (int i = 0; i < 8; ++i) t += red[i];
        partial[(blockIdx.z * gridDim.y + blockIdx.y) * gridDim.x + blockIdx.x] = t;
    }
}

__global__ void structure_loss_finalize(const float* __restrict__ partial,
                                        float* __restrict__ out) {
    __shared__ float red[256];
    float a = 0.0f;
    for (int i = threadIdx.x; i < NBLOCKS; i += 256) a += partial[i];
    red[threadIdx.x] = a;
    __syncthreads();
    for (int s = 128; s > 0; s >>= 1) {
        if (threadIdx.x < s) red[threadIdx.x] += red[threadIdx.x + s];
        __syncthreads();
    }
    if (threadIdx.x == 0) out[0] = 1.0f - red[0] / NVALID;
}

extern "C" void kernel_launch(void* const* d_in, const int* in_sizes, int n_in,
                              void* d_out, int out_size, void* d_ws, size_t ws_size,
                              hipStream_t stream) {
    const float* x = (const float*)d_in[0];   // (2,4,160,160,96) f32
    const int*   y = (const int*)d_in[1];     // (2,1,160,160,96) int
    float* partial = (float*)d_ws;            // NBLOCKS floats of scratch
    float* out     = (float*)d_out;           // scalar loss

    dim3 grid(DT_TILES * WT_TILES, HT_TILES, 2);
    structure_loss_tile<<<grid, 256, 0, stream>>>(x, y, partial);
    structure_loss_finalize<<<1, 256, 0, stream>>>(partial, out);
}